// SSIM_87024627351738
// MI455X (gfx1250) — compile-verified
//
#include <hip/hip_runtime.h>
#include <math.h>

typedef __attribute__((ext_vector_type(2))) float v2f;
typedef __attribute__((ext_vector_type(8))) float v8f;

#define IMG_H 512
#define IMG_W 512
#define PLANES 48           // 16 * 3
#define TILE 32
#define EXT 42              // TILE + 10 halo
#define EXT_ROWS 48         // padded to 3*16 for tiling
#define EXT_STRIDE 44       // covers K overreach (xb=16 + k=27 -> 43); bank-friendly
#define HT_ROWS 48          // 3*16: no store guard needed (reads only reach row 43)
#define HT_STRIDE 42        // 2*42 % 64 = 20, 8*42 % 64 = 16 -> conflict-free
#define SSIM_C1 1e-4f
#define SSIM_C2 9e-4f
#define N_TOTAL 12582912.0f // 16*3*512*512
#define TILES_PER_PLANE 256 // (512/32)^2
#define NBLOCKS (PLANES * TILES_PER_PLANE)

// Horizontal-blur of one 16x16 tile of field F via chained f32 WMMA.
// F: 0 -> I1, 1 -> I2, 2 -> I1^2, 3 -> I2^2, 4 -> I1*I2 (compile-time).
template <int F>
__device__ __forceinline__ v8f hpass_tile(
    const float (*__restrict__ e1)[EXT_STRIDE],
    const float (*__restrict__ e2)[EXT_STRIDE],
    int rowbase, int xb, int half, int mn,
    const float* __restrict__ wx, const float* __restrict__ wy) {
  v8f acc = {0.f, 0.f, 0.f, 0.f, 0.f, 0.f, 0.f, 0.f};
  const int row = rowbase + mn;  // A: lane = M
#pragma unroll
  for (int kk = 0; kk < 7; ++kk) {
    const int ja = xb + kk * 4 + 2 * half;  // A: K striped over VGPR + lane-half
    float fa, fb;
    if (F == 0) {
      fa = e1[row][ja];
      fb = e1[row][ja + 1];
    } else if (F == 1) {
      fa = e2[row][ja];
      fb = e2[row][ja + 1];
    } else if (F == 2) {
      float a = e1[row][ja], b = e1[row][ja + 1];
      fa = a * a; fb = b * b;
    } else if (F == 3) {
      float a = e2[row][ja], b = e2[row][ja + 1];
      fa = a * a; fb = b * b;
    } else {
      float a1 = e1[row][ja], b1 = e1[row][ja + 1];
      float a2 = e2[row][ja], b2 = e2[row][ja + 1];
      fa = a1 * a2; fb = b1 * b2;
    }
    v2f A = {fa, fb};
    v2f B = {wx[kk], wy[kk]};  // Toeplitz band (K x N, lane = N)
    acc = __builtin_amdgcn_wmma_f32_16x16x4_f32(
        false, A, false, B, (short)0, acc, false, false);
  }
  return acc;
}

__global__ __launch_bounds__(128) void ssim_tile_kernel(
    const float* __restrict__ img1, const float* __restrict__ img2,
    float* __restrict__ partial) {
  __shared__ float ext1[EXT_ROWS][EXT_STRIDE];
  __shared__ float ext2[EXT_ROWS][EXT_STRIDE];
  __shared__ float htmp[5][HT_ROWS][HT_STRIDE];
  __shared__ float gw[11];
  __shared__ float wsum[4];

  const int tid  = threadIdx.x;
  // readfirstlane makes the wave index compiler-visible uniform -> scalar
  // branches (no EXEC masking around the WMMAs).
  const int wave = __builtin_amdgcn_readfirstlane(tid >> 5);
  const int lane = tid & 31;
  const int half = lane >> 4;   // which 16-lane half of the wave
  const int mn   = lane & 15;   // M (A row) or N (B/C/D column)

  const int t     = blockIdx.x % TILES_PER_PLANE;
  const int plane = blockIdx.x / TILES_PER_PLANE;
  const int py0 = (t / (IMG_W / TILE)) * TILE;
  const int px0 = (t % (IMG_W / TILE)) * TILE;
  const float* p1 = img1 + (size_t)plane * IMG_H * IMG_W;
  const float* p2 = img2 + (size_t)plane * IMG_H * IMG_W;

  // normalized 11-tap gaussian (sigma=1.5), same math as reference
  if (tid == 0) {
    float tmp[11]; float s = 0.f;
    for (int i = 0; i < 11; ++i) {
      float x = (float)(i - 5);
      tmp[i] = expf(-(x * x) / 4.5f);
      s += tmp[i];
    }
    for (int i = 0; i < 11; ++i) gw[i] = tmp[i] / s;
  }

  // stage 1: halo tiles into LDS, zero outside image and in pad region
  for (int idx = tid; idx < EXT_ROWS * EXT_STRIDE; idx += 128) {
    int r = idx / EXT_STRIDE, c = idx % EXT_STRIDE;
    float v1 = 0.f, v2 = 0.f;
    if (r < EXT && c < EXT) {
      int gy = py0 - 5 + r, gx = px0 - 5 + c;
      if (gy >= 0 && gy < IMG_H && gx >= 0 && gx < IMG_W) {
        v1 = p1[gy * IMG_W + gx];
        v2 = p2[gy * IMG_W + gx];
      }
    }
    ext1[r][c] = v1;
    ext2[r][c] = v2;
  }
  __syncthreads();

  // Per-lane Toeplitz weights, one pair per K=4 step. By symmetry the same
  // values serve as B (K x N, lane = N) in pass 1 and A (M x K, lane = M)
  // in pass 2: value = g[k - mn] for k = kk*4 + 2*half (+1), 0 off-band.
  float wx[7], wy[7];
#pragma unroll
  for (int kk = 0; kk < 7; ++kk) {
    int d = kk * 4 + 2 * half - mn;
    wx[kk] = ((unsigned)d <= 10u) ? gw[d] : 0.f;
    int d1 = d + 1;
    wy[kk] = ((unsigned)d1 <= 10u) ? gw[d1] : 0.f;
  }

  // stage 2: horizontal blur of 5 fields via WMMA.
  // 5 fields x 3 row-tiles x 2 col-tiles = 30 wave-tiles, round-robin by wave.
  for (int tt = wave; tt < 30; tt += 4) {
    const int f  = tt / 6;        // uniform (tt uniform)
    const int rr = tt - f * 6;
    const int rowbase = (rr >> 1) * 16;
    const int xb = (rr & 1) * 16;
    v8f acc;
    switch (f) {  // uniform switch -> scalar branches, EXEC stays all-1s
      case 0:  acc = hpass_tile<0>(ext1, ext2, rowbase, xb, half, mn, wx, wy); break;
      case 1:  acc = hpass_tile<1>(ext1, ext2, rowbase, xb, half, mn, wx, wy); break;
      case 2:  acc = hpass_tile<2>(ext1, ext2, rowbase, xb, half, mn, wx, wy); break;
      case 3:  acc = hpass_tile<3>(ext1, ext2, rowbase, xb, half, mn, wx, wy); break;
      default: acc = hpass_tile<4>(ext1, ext2, rowbase, xb, half, mn, wx, wy); break;
    }
    // D layout: VGPR v -> M = v + 8*half, lane -> N. HT_ROWS=48 -> no guard.
#pragma unroll
    for (int v = 0; v < 8; ++v)
      htmp[f][rowbase + v + 8 * half][xb + mn] = acc[v];
  }
  __syncthreads();

  // stage 3: vertical blur (A = Toeplitz, B = htmp columns) + SSIM map.
  // One 16x16 output tile per wave.
  const int ybase = (wave >> 1) * 16;   // uniform
  const int x0t   = (wave & 1) * 16;    // uniform
  v8f accs[5];
#pragma unroll
  for (int f = 0; f < 5; ++f) {
    v8f acc = {0.f, 0.f, 0.f, 0.f, 0.f, 0.f, 0.f, 0.f};
#pragma unroll
    for (int kk = 0; kk < 7; ++kk) {
      int rb = ybase + kk * 4 + 2 * half;  // B: lane = N, K striped over VGPR+half
      v2f A = {wx[kk], wy[kk]};
      v2f B = {htmp[f][rb][x0t + mn], htmp[f][rb + 1][x0t + mn]};
      acc = __builtin_amdgcn_wmma_f32_16x16x4_f32(
          false, A, false, B, (short)0, acc, false, false);
    }
    accs[f] = acc;
  }

  float lsum = 0.f;
#pragma unroll
  for (int v = 0; v < 8; ++v) {
    float mu1 = accs[0][v], mu2 = accs[1][v];
    float b11 = accs[2][v], b22 = accs[3][v], b12 = accs[4][v];
    float mu1sq = mu1 * mu1, mu2sq = mu2 * mu2, mu12 = mu1 * mu2;
    float s1 = b11 - mu1sq;
    float s2 = b22 - mu2sq;
    float s12 = b12 - mu12;
    float num = (2.f * mu12 + SSIM_C1) * (2.f * s12 + SSIM_C2);
    float den = (mu1sq + mu2sq + SSIM_C1) * (s1 + s2 + SSIM_C2);
    lsum += num / den;
  }
  // deterministic wave reduction
#pragma unroll
  for (int off = 16; off > 0; off >>= 1) lsum += __shfl_xor(lsum, off, 32);
  if (lane == 0) wsum[wave] = lsum;
  __syncthreads();
  if (tid == 0) partial[blockIdx.x] = (wsum[0] + wsum[1]) + (wsum[2] + wsum[3]);
}

__global__ __launch_bounds__(256) void ssim_reduce_kernel(
    const float* __restrict__ partial, float* __restrict__ out, int n) {
  __shared__ float s[256];
  int tid = threadIdx.x;
  float v = 0.f;
  for (int i = tid; i < n; i += 256) v += partial[i];  // fixed order: deterministic
  s[tid] = v;
  __syncthreads();
  for (int st = 128; st > 0; st >>= 1) {
    if (tid < st) s[tid] += s[tid + st];
    __syncthreads();
  }
  if (tid == 0) out[0] = s[0] * (1.0f / N_TOTAL);
}

extern "C" void kernel_launch(void* const* d_in, const int* in_sizes, int n_in,
                              void* d_out, int out_size, void* d_ws, size_t ws_size,
                              hipStream_t stream) {
  const float* img1 = (const float*)d_in[0];
  const float* img2 = (const float*)d_in[1];
  float* out = (float*)d_out;
  float* partial = (float*)d_ws;  // NBLOCKS floats = 48 KiB of scratch

  ssim_tile_kernel<<<NBLOCKS, 128, 0, stream>>>(img1, img2, partial);
  ssim_reduce_kernel<<<1, 256, 0, stream>>>(partial, out, NBLOCKS);
}